// DWT2D_11802570129596
// MI455X (gfx1250) — compile-verified
//
#include <hip/hip_runtime.h>
#include <hip/hip_bf16.h>

// Haar DWT2 (8,32,512,512) f32 -> 4x (8,32,256,256) f32.
// Memory-bound streaming kernel; the 4->4 Haar butterfly rides on
// V_WMMA_F32_16X16X4_F32 with the analysis matrix duplicated into rows
// {0,1} (LL,LH) and {8,9} (HL,HH) so that BOTH lane-halves of the wave
// receive useful D values -> branchless full-wave stores.

typedef float v2f __attribute__((ext_vector_type(2)));
typedef float v8f __attribute__((ext_vector_type(8)));

__global__ __launch_bounds__(256) void dwt2_haar_wmma(const float* __restrict__ x,
                                                      float* __restrict__ out) {
    const int lane = threadIdx.x & 31;   // wave32
    const int wave = threadIdx.x >> 5;   // 8 waves per workgroup
    const int m    = lane & 15;
    const int half = lane >> 4;          // 0: even image row / K{0,1}   1: odd row / K{2,3}

    // wave id -> (image bc, block-row i). 256 images * 256 block-rows = 65536 waves.
    const unsigned w  = blockIdx.x * 8u + (unsigned)wave;
    const unsigned bc = w >> 8;
    const unsigned i  = w & 255u;

    // ---- A matrix (16x4): Haar rows duplicated so both lane halves of D are live.
    //   row 0 = LL [+,+,+,+]   row 1 = LH [+,+,-,-]
    //   row 8 = HL [+,-,+,-]   row 9 = HH [+,-,-,+]   (all other rows zero)
    // Layout: lane M holds (K0,K1), lane M+16 holds (K2,K3) for row M = lane&15.
    v2f a;
    {
        const bool valid = (m & 6) == 0;                 // M in {0,1,8,9}
        const int  r     = (m & 1) | ((m >> 3) << 1);    // Haar row index 0..3
        const int  k0    = 2 * half, k1 = k0 + 1;
        const int  s0 = ((r & 1) & ((k0 >> 1) & 1)) ^ (((r >> 1) & 1) & (k0 & 1));
        const int  s1 = ((r & 1) & ((k1 >> 1) & 1)) ^ (((r >> 1) & 1) & (k1 & 1));
        a.x = valid ? (s0 ? -0.5f : 0.5f) : 0.0f;
        a.y = valid ? (s1 ? -0.5f : 0.5f) : 0.0f;
    }

    // ---- B loads: lane n<16 reads (x00,x01) of block n from the even row,
    //      lane 16+n reads (x10,x11) of block n from the odd row.
    // Each half-wave covers one contiguous 128B segment -> fully coalesced b64s.
    const float* src = x + ((size_t)(bc * 512u + 2u * i + (unsigned)half)) * 512u
                         + (size_t)(2 * m);

    // ---- Store pointers (computed once; loop uses immediate offsets).
    // lanes 0-15:  d[0]->LL plane, d[1]->LH plane
    // lanes 16-31: d[0]->HL plane, d[1]->HH plane
    const size_t P     = 16777216;  // plane stride in elements
    const size_t obase = ((size_t)bc * 256u + i) * 256u + (size_t)m;
    float* outA = out + (size_t)(half ? 2u : 0u) * P + obase;
    float* outB = out + (size_t)(half ? 3u : 1u) * P + obase;

    const v8f czero = {};

    // ---- Software pipeline: clause of 16 NT b64 loads, then 16 wmma+store stages.
    v2f bv[16];
#pragma unroll
    for (int t = 0; t < 16; ++t)
        bv[t] = __builtin_nontemporal_load((const v2f*)(src + 32 * t));

#pragma unroll
    for (int t = 0; t < 16; ++t) {
        v8f d = __builtin_amdgcn_wmma_f32_16x16x4_f32(
            /*neg_a=*/false, a, /*neg_b=*/false, bv[t],
            /*c_mod=*/(short)0, czero, /*reuse_a=*/false, /*reuse_b=*/false);
        __builtin_nontemporal_store(d[0], outA + 16 * t);
        __builtin_nontemporal_store(d[1], outB + 16 * t);
    }
}

extern "C" void kernel_launch(void* const* d_in, const int* in_sizes, int n_in,
                              void* d_out, int out_size, void* d_ws, size_t ws_size,
                              hipStream_t stream) {
    (void)in_sizes; (void)n_in; (void)d_ws; (void)ws_size; (void)out_size;
    const float* x = (const float*)d_in[0];
    float* out = (float*)d_out;
    // 65536 block-rows, 8 waves (one row each) per 256-thread workgroup.
    dim3 grid(8192), block(256);
    dwt2_haar_wmma<<<grid, block, 0, stream>>>(x, out);
}